// CombinedCS_9500467658912
// MI455X (gfx1250) — compile-verified
//
#include <hip/hip_runtime.h>
#include <math.h>

// ---------------------------------------------------------------------------
// Problem constants (match reference)
// ---------------------------------------------------------------------------
#define NSIDE   320
#define HWSZ    (NSIDE*NSIDE)        // 102400
#define NB      16
#define NCH     2
#define FIELD   (NB*NCH*HWSZ)        // 3,276,800 floats per (B,2,H,W) field
#define NPL     (NB*NCH)             // 32 independent image planes
#define LAM_TV  0.005f
#define LAM_WAV 0.005f
#define TAU_TV  0.25f
#define MAX_ITER 15
#define TV_ITERS 5
#define KC      16                   // K-chunk staged in LDS per GEMM step

typedef float v2f __attribute__((ext_vector_type(2)));
typedef float v8f __attribute__((ext_vector_type(8)));
typedef int   v4i __attribute__((ext_vector_type(4)));

// ---------------------------------------------------------------------------
// CDNA5 async global->LDS copy (ASYNCcnt path) with safe fallback
// Builtin signature (probe-confirmed by clang diagnostic):
//   void __builtin_amdgcn_global_load_async_to_lds_b128(
//        v4i addrspace(1)* gsrc, v4i addrspace(3)* ldst, imm offset, imm cpol)
// ---------------------------------------------------------------------------
#if defined(__has_builtin)
#if __has_builtin(__builtin_amdgcn_global_load_async_to_lds_b128)
#define HAVE_ASYNC_LDS 1
#endif
#endif

#ifdef HAVE_ASYNC_LDS
static __device__ inline void cp16_async(float* lds, const float* g) {
  __builtin_amdgcn_global_load_async_to_lds_b128(
      (__attribute__((address_space(1))) v4i*)g,
      (__attribute__((address_space(3))) v4i*)lds,
      /*offset=*/0, /*cpol=*/0);
}
static __device__ inline void wait_async0() {
#if __has_builtin(__builtin_amdgcn_s_wait_asynccnt)
  __builtin_amdgcn_s_wait_asynccnt(0);
#else
  asm volatile("s_wait_asynccnt 0x0" ::: "memory");
#endif
}
#else
static __device__ inline void cp16_async(float* lds, const float* g) {
  float4 v = *(const float4*)g;       // global_load_b128
  *(float4*)lds = v;                  // ds_store_b128
}
static __device__ inline void wait_async0() {}
#endif

// ---------------------------------------------------------------------------
// fp32 WMMA 16x16x4 (CDNA5: V_WMMA_F32_16X16X4_F32)
// ---------------------------------------------------------------------------
static __device__ inline v8f wmma4(v2f a, v2f b, v8f c) {
  return __builtin_amdgcn_wmma_f32_16x16x4_f32(false, a, false, b, (short)0, c,
                                               false, false);
}

// ---------------------------------------------------------------------------
// Build shift-baked unitary DFT matrices:
//   Ff[u,n] = exp(-2*pi*i*s(u)*s(n)/N)/sqrt(N),  s(k) = (k+N/2)%N
//   Fi      = conj(Ff);  fftshift(fft(ifftshift(x))) == Ff @ x; Ff symmetric.
// ---------------------------------------------------------------------------
__global__ __launch_bounds__(256) void build_dft_k(float* __restrict__ Ffre,
                                                   float* __restrict__ Ffim,
                                                   float* __restrict__ Fire,
                                                   float* __restrict__ Fiim) {
  int tid = blockIdx.x * blockDim.x + threadIdx.x;
  if (tid >= HWSZ) return;
  int u = tid / NSIDE, n = tid % NSIDE;
  int su = (u + NSIDE / 2) % NSIDE;
  int sn = (n + NSIDE / 2) % NSIDE;
  int p = (su * sn) % NSIDE;
  float ang = (2.0f * 3.14159265358979323846f * (float)p) / (float)NSIDE;
  float inv = rsqrtf((float)NSIDE);
  float c = cosf(ang) * inv;
  float s = sinf(ang) * inv;
  Ffre[tid] = c;  Ffim[tid] = -s;
  Fire[tid] = c;  Fiim[tid] = +s;
}

// ---------------------------------------------------------------------------
// Complex GEMM: C = A*B, 320x320 row-major, batched over blockIdx.z.
// Block tile 64x64 (complex), 8 waves in a 2x4 grid, each wave owns two
// vertically-stacked 16x16 C tiles (8 f32 accumulators).
// K is consumed in LDS-staged chunks of KC=16 (async global->LDS on CDNA5).
//
// A fragment (16x4 f32): lane L holds row M=L%16; v[j] = A[M][k+2*(L/16)+j]
// B fragment (4x16 f32): lane L holds col N=L%16; v[j] = B[k+2*j+(L/16)][N]
// C/D (16x16 f32): v[r] -> C[m0 + r + 8*(L/16)][n0 + L%16]
// ---------------------------------------------------------------------------
__global__ __launch_bounds__(256) void gemm320_c(
    const float* __restrict__ Are, const float* __restrict__ Aim, size_t aStr,
    const float* __restrict__ Bre, const float* __restrict__ Bim, size_t bStr,
    float* __restrict__ Cre, float* __restrict__ Cim, size_t cStr) {
  const int N = NSIDE;
  __shared__ float Asre[64 * KC], Asim[64 * KC];
  __shared__ float Bsre[KC * 64], Bsim[KC * 64];

  size_t bz = blockIdx.z;
  Are += bz * aStr;  Aim += bz * aStr;
  Bre += bz * bStr;  Bim += bz * bStr;
  Cre += bz * cStr;  Cim += bz * cStr;

  const int tid = threadIdx.x;
  const int lane = tid & 31, wave = tid >> 5;
  const int wc = wave & 3, wr = wave >> 2;            // 2x4 wave grid
  const int M0 = blockIdx.x * 64, N0 = blockIdx.y * 64;

  // staging assignment: A tile 64x16, B tile 16x64, one float4 per thread each
  const int arow = tid >> 2, aoff = (tid & 3) * 4;
  const int brow = tid >> 4, boff = (tid & 15) * 4;

  const int mr = lane & 15, ksel = lane >> 4;
  const int ncl = wc * 16 + (lane & 15);              // local column

  v8f acc[2][4];
#pragma unroll
  for (int t = 0; t < 2; ++t)
#pragma unroll
    for (int q = 0; q < 4; ++q) acc[t][q] = (v8f){};

  for (int k0 = 0; k0 < N; k0 += KC) {
    __syncthreads();   // previous chunk's LDS readers done
    cp16_async(&Asre[arow * KC + aoff], Are + (size_t)(M0 + arow) * N + k0 + aoff);
    cp16_async(&Asim[arow * KC + aoff], Aim + (size_t)(M0 + arow) * N + k0 + aoff);
    cp16_async(&Bsre[brow * 64 + boff], Bre + (size_t)(k0 + brow) * N + N0 + boff);
    cp16_async(&Bsim[brow * 64 + boff], Bim + (size_t)(k0 + brow) * N + N0 + boff);
    wait_async0();
    __syncthreads();

#pragma unroll
    for (int kk = 0; kk < KC; kk += 4) {
      v2f bre = {Bsre[(kk + ksel) * 64 + ncl], Bsre[(kk + 2 + ksel) * 64 + ncl]};
      v2f bim = {Bsim[(kk + ksel) * 64 + ncl], Bsim[(kk + 2 + ksel) * 64 + ncl]};
#pragma unroll
      for (int t = 0; t < 2; ++t) {
        int rl = wr * 32 + t * 16 + mr;
        v2f are = {Asre[rl * KC + kk + 2 * ksel], Asre[rl * KC + kk + 2 * ksel + 1]};
        v2f aim = {Asim[rl * KC + kk + 2 * ksel], Asim[rl * KC + kk + 2 * ksel + 1]};
        acc[t][0] = wmma4(are, bre, acc[t][0]);   // Re*Re
        acc[t][1] = wmma4(aim, bim, acc[t][1]);   // Im*Im
        acc[t][2] = wmma4(are, bim, acc[t][2]);   // Re*Im
        acc[t][3] = wmma4(aim, bre, acc[t][3]);   // Im*Re
      }
    }
  }

  const int nout = N0 + ncl;
#pragma unroll
  for (int t = 0; t < 2; ++t) {
    int mbase = M0 + wr * 32 + t * 16 + ((lane >> 4) << 3);
#pragma unroll
    for (int r = 0; r < 8; ++r) {
      size_t idx = (size_t)(mbase + r) * N + nout;
      Cre[idx] = acc[t][0][r] - acc[t][1][r];
      Cim[idx] = acc[t][2][r] + acc[t][3][r];
    }
  }
}

// ---------------------------------------------------------------------------
// Elementwise kernels
// ---------------------------------------------------------------------------
__global__ __launch_bounds__(256) void ew_mask_k(float* __restrict__ K,
                                                 const float* __restrict__ y,
                                                 const float* __restrict__ mask) {
  int tid = blockIdx.x * blockDim.x + threadIdx.x;   // over B*HW
  int b = tid / HWSZ, hw = tid % HWSZ;
  float m = mask[(size_t)b * HWSZ + hw];
  size_t i0 = (size_t)b * NCH * HWSZ + hw;
  K[i0]        = m * (m * K[i0]        - y[i0]);
  K[i0 + HWSZ] = m * (m * K[i0 + HWSZ] - y[i0 + HWSZ]);
}

__global__ __launch_bounds__(256) void ew_zstep_k(float* __restrict__ g,
                                                  const float* __restrict__ z) {
  int tid = blockIdx.x * blockDim.x + threadIdx.x;
  g[tid] = z[tid] - g[tid];
}

__global__ __launch_bounds__(256) void zero_k(float* __restrict__ p) {
  int tid = blockIdx.x * blockDim.x + threadIdx.x;
  p[tid] = 0.0f;
}

// ---------------------------------------------------------------------------
// TV prox (Chambolle dual iterations), planes of 320x320, p = (NPL,2,H,W)
// ---------------------------------------------------------------------------
static __device__ inline float divp_at(const float* __restrict__ px,
                                       const float* __restrict__ py,
                                       int h, int w) {
  const int W = NSIDE, H = NSIDE;
  float dx;
  if (w == 0)            dx = px[(size_t)h * W];
  else if (w == W - 1)   dx = -px[(size_t)h * W + (W - 2)];
  else                   dx = px[(size_t)h * W + w] - px[(size_t)h * W + w - 1];
  float dy;
  if (h == 0)            dy = py[w];
  else if (h == H - 1)   dy = -py[(size_t)(H - 2) * W + w];
  else                   dy = py[(size_t)h * W + w] - py[(size_t)(h - 1) * W + w];
  return dx + dy;
}

__global__ __launch_bounds__(256) void tv_iter_k(const float* __restrict__ xc,
                                                 const float* __restrict__ Pin,
                                                 float* __restrict__ Pout) {
  const int W = NSIDE, H = NSIDE;
  int tid = blockIdx.x * blockDim.x + threadIdx.x;   // over NPL*HW
  int bc = tid / HWSZ, rem = tid % HWSZ;
  int h = rem / W, w = rem % W;
  const float* x = xc + (size_t)bc * HWSZ;
  const float* px = Pin + (size_t)bc * 2 * HWSZ;
  const float* py = px + HWSZ;
  float u00 = x[(size_t)h * W + w] - LAM_TV * divp_at(px, py, h, w);
  float gx = 0.0f, gy = 0.0f;
  if (w < W - 1) {
    float u01 = x[(size_t)h * W + w + 1] - LAM_TV * divp_at(px, py, h, w + 1);
    gx = u01 - u00;
  }
  if (h < H - 1) {
    float u10 = x[(size_t)(h + 1) * W + w] - LAM_TV * divp_at(px, py, h + 1, w);
    gy = u10 - u00;
  }
  float pnx = px[(size_t)h * W + w] + TAU_TV * gx;
  float pny = py[(size_t)h * W + w] + TAU_TV * gy;
  float nrm = sqrtf(pnx * pnx + pny * pny + 1e-8f);
  float s = 1.0f / fmaxf(nrm, 1.0f);
  float* ox = Pout + (size_t)bc * 2 * HWSZ;
  float* oy = ox + HWSZ;
  ox[(size_t)h * W + w] = pnx * s;
  oy[(size_t)h * W + w] = pny * s;
}

__global__ __launch_bounds__(256) void tv_final_k(float* __restrict__ xc,
                                                  const float* __restrict__ P) {
  const int W = NSIDE;
  int tid = blockIdx.x * blockDim.x + threadIdx.x;
  int bc = tid / HWSZ, rem = tid % HWSZ;
  int h = rem / W, w = rem % W;
  const float* px = P + (size_t)bc * 2 * HWSZ;
  const float* py = px + HWSZ;
  xc[tid] = xc[tid] - LAM_TV * divp_at(px, py, h, w);
}

// ---------------------------------------------------------------------------
// Haar DWT with soft-thresholded details / IDWT
// ---------------------------------------------------------------------------
static __device__ inline float softf(float v, float t) {
  return copysignf(fmaxf(fabsf(v) - t, 0.0f), v);
}

__global__ __launch_bounds__(256) void dwt_soft_k(const float* __restrict__ src,
                                                  float* __restrict__ ll,
                                                  float* __restrict__ dd,
                                                  int ho, int wo) {
  int tid = blockIdx.x * blockDim.x + threadIdx.x;
  int hw = ho * wo;
  int p = tid / hw, rem = tid % hw;
  int i = rem / wo, j = rem % wo;
  int ws2 = 2 * wo;
  const float* sp = src + (size_t)p * (4 * hw);
  float a = sp[(size_t)(2 * i) * ws2 + 2 * j];
  float b = sp[(size_t)(2 * i) * ws2 + 2 * j + 1];
  float c = sp[(size_t)(2 * i + 1) * ws2 + 2 * j];
  float d = sp[(size_t)(2 * i + 1) * ws2 + 2 * j + 1];
  float llv = 0.5f * (a + b + c + d);
  float lh  = 0.5f * (a + b - c - d);
  float hl  = 0.5f * (a - b + c - d);
  float hh  = 0.5f * (a - b - c + d);
  ll[(size_t)p * hw + rem] = llv;
  dd[((size_t)p * 3 + 0) * hw + rem] = softf(lh, LAM_WAV);
  dd[((size_t)p * 3 + 1) * hw + rem] = softf(hl, LAM_WAV);
  dd[((size_t)p * 3 + 2) * hw + rem] = softf(hh, LAM_WAV);
}

__global__ __launch_bounds__(256) void idwt_k(const float* __restrict__ ll,
                                              const float* __restrict__ dd,
                                              float* __restrict__ dst,
                                              int ho, int wo) {
  int tid = blockIdx.x * blockDim.x + threadIdx.x;
  int hw = ho * wo;
  int p = tid / hw, rem = tid % hw;
  int i = rem / wo, j = rem % wo;
  int ws2 = 2 * wo;
  float llv = ll[(size_t)p * hw + rem];
  float lh  = dd[((size_t)p * 3 + 0) * hw + rem];
  float hl  = dd[((size_t)p * 3 + 1) * hw + rem];
  float hh  = dd[((size_t)p * 3 + 2) * hw + rem];
  float* dp = dst + (size_t)p * (4 * hw);
  dp[(size_t)(2 * i) * ws2 + 2 * j]         = 0.5f * (llv + lh + hl + hh);
  dp[(size_t)(2 * i) * ws2 + 2 * j + 1]     = 0.5f * (llv + lh - hl - hh);
  dp[(size_t)(2 * i + 1) * ws2 + 2 * j]     = 0.5f * (llv - lh + hl - hh);
  dp[(size_t)(2 * i + 1) * ws2 + 2 * j + 1] = 0.5f * (llv - lh - hl + hh);
}

// FISTA momentum: z = xn + m*(xn - x); x = xn
__global__ __launch_bounds__(256) void fista_k(const float* __restrict__ xn,
                                               float* __restrict__ x,
                                               float* __restrict__ z,
                                               float m) {
  int tid = blockIdx.x * blockDim.x + threadIdx.x;
  float v = xn[tid];
  float xo = x[tid];
  x[tid] = v;
  z[tid] = v + m * (v - xo);
}

// ---------------------------------------------------------------------------
// Host orchestration
// ---------------------------------------------------------------------------
static void gemmC(const float* Ar, const float* Ai, size_t as,
                  const float* Br, const float* Bi, size_t bs,
                  float* Cr, float* Ci, size_t cs, hipStream_t s) {
  dim3 g(NSIDE / 64, NSIDE / 64, NB);   // (5,5,16)
  gemm320_c<<<g, 256, 0, s>>>(Ar, Ai, as, Br, Bi, bs, Cr, Ci, cs);
}

extern "C" void kernel_launch(void* const* d_in, const int* in_sizes, int n_in,
                              void* d_out, int out_size, void* d_ws, size_t ws_size,
                              hipStream_t stream) {
  (void)in_sizes; (void)n_in; (void)out_size; (void)ws_size;
  const float* y    = (const float*)d_in[0];   // (16,2,320,320)
  const float* mask = (const float*)d_in[1];   // (16,1,320,320)

  float* w = (float*)d_ws;
  float* Ffre = w; w += HWSZ;
  float* Ffim = w; w += HWSZ;
  float* Fire = w; w += HWSZ;
  float* Fiim = w; w += HWSZ;
  float* X  = w; w += FIELD;
  float* Z  = w; w += FIELD;
  float* T1 = w; w += FIELD;
  float* T2 = w; w += FIELD;
  float* P0 = w; w += 2 * (size_t)FIELD;
  float* P1 = w; w += 2 * (size_t)FIELD;
  float* LL0 = w; w += (size_t)NPL * 160 * 160;
  float* D0  = w; w += (size_t)NPL * 3 * 160 * 160;
  float* LL1 = w; w += (size_t)NPL * 80 * 80;
  float* D1  = w; w += (size_t)NPL * 3 * 80 * 80;
  float* LL2 = w; w += (size_t)NPL * 40 * 40;
  float* D2  = w; w += (size_t)NPL * 3 * 40 * 40;

  const size_t FSTR = (size_t)NCH * HWSZ;   // per-batch stride in a field

  build_dft_k<<<HWSZ / 256, 256, 0, stream>>>(Ffre, Ffim, Fire, Fiim);

  // x0 = ifft2c(y):  T1 = Y*Fi ; X = Fi*T1
  gemmC(y, y + HWSZ, FSTR, Fire, Fiim, 0, T1, T1 + HWSZ, FSTR, stream);
  gemmC(Fire, Fiim, 0, T1, T1 + HWSZ, FSTR, X, X + HWSZ, FSTR, stream);
  (void)hipMemcpyAsync(Z, X, (size_t)FIELD * sizeof(float),
                       hipMemcpyDeviceToDevice, stream);

  float t = 1.0f;
  for (int it = 0; it < MAX_ITER; ++it) {
    // K = fft2c(z) -> T2
    gemmC(Z, Z + HWSZ, FSTR, Ffre, Ffim, 0, T1, T1 + HWSZ, FSTR, stream);
    gemmC(Ffre, Ffim, 0, T1, T1 + HWSZ, FSTR, T2, T2 + HWSZ, FSTR, stream);
    // T2 = mask*(mask*T2 - y)
    ew_mask_k<<<(NB * HWSZ) / 256, 256, 0, stream>>>(T2, y, mask);
    // g = ifft2c(T2) -> T2
    gemmC(T2, T2 + HWSZ, FSTR, Fire, Fiim, 0, T1, T1 + HWSZ, FSTR, stream);
    gemmC(Fire, Fiim, 0, T1, T1 + HWSZ, FSTR, T2, T2 + HWSZ, FSTR, stream);
    // z_step = z - g  (into T2)
    ew_zstep_k<<<FIELD / 256, 256, 0, stream>>>(T2, Z);

    // prox_tv on T2 (p ping-pong, p0 = 0)
    zero_k<<<(2 * FIELD) / 256, 256, 0, stream>>>(P0);
    float* pin = P0; float* pout = P1;
    for (int tv = 0; tv < TV_ITERS; ++tv) {
      tv_iter_k<<<(NPL * HWSZ) / 256, 256, 0, stream>>>(T2, pin, pout);
      float* tmp = pin; pin = pout; pout = tmp;
    }
    tv_final_k<<<(NPL * HWSZ) / 256, 256, 0, stream>>>(T2, pin);

    // 3-level Haar with soft-thresholded details, then reconstruct -> T1
    dwt_soft_k<<<(NPL * 160 * 160) / 256, 256, 0, stream>>>(T2, LL0, D0, 160, 160);
    dwt_soft_k<<<(NPL * 80 * 80) / 256, 256, 0, stream>>>(LL0, LL1, D1, 80, 80);
    dwt_soft_k<<<(NPL * 40 * 40) / 256, 256, 0, stream>>>(LL1, LL2, D2, 40, 40);
    idwt_k<<<(NPL * 40 * 40) / 256, 256, 0, stream>>>(LL2, D2, LL1, 40, 40);
    idwt_k<<<(NPL * 80 * 80) / 256, 256, 0, stream>>>(LL1, D1, LL0, 80, 80);
    idwt_k<<<(NPL * 160 * 160) / 256, 256, 0, stream>>>(LL0, D0, T1, 160, 160);

    // FISTA momentum update (t recursion on host; deterministic)
    float tn = (1.0f + sqrtf(1.0f + 4.0f * t * t)) * 0.5f;
    float mcoef = (t - 1.0f) / tn;
    fista_k<<<FIELD / 256, 256, 0, stream>>>(T1, X, Z, mcoef);
    t = tn;
  }

  (void)hipMemcpyAsync(d_out, X, (size_t)FIELD * sizeof(float),
                       hipMemcpyDeviceToDevice, stream);
}